// FilterBank_79362405695973
// MI455X (gfx1250) — compile-verified
//
#include <hip/hip_runtime.h>

// ---------------------------------------------------------------------------
// filtfilt 9-band IIR filter bank on MI455X (gfx1250), WMMA-based.
//
// Reference: y = lfilter(x); y = reverse(lfilter(reverse(y))) per band,
// order-8 transposed direct-form II, x:(64,32,4096) f32 -> out:(64,9,32,4096).
//
// Strategy: block state-space. For block length L=16:
//     [ Y(16xN) ]   [ Phi  T   ] [ Z (8xN)  ]
//     [ Z'(8xN) ] = [ A^16 Gam ] [ X (16xN) ]   == M(24x24) * U(24xN), N=16 seqs
// Contract the 24-deep K dimension with six chained V_WMMA_F32_16X16X4_F32
// per output tile (full f32 precision -- IIR state feedback is not an
// f16/fp8-tolerant computation). State-dependent K-slices are contracted LAST
// so the serial cross-block WMMA chain is only 2 deep.
// ---------------------------------------------------------------------------

typedef __attribute__((ext_vector_type(2))) float v2f;
typedef __attribute__((ext_vector_type(8))) float v8f;

#define BANDS          9
#define NCOEF          9      // order 8 -> 9 coefficients
#define ORDER          8
#define T_LEN          4096
#define L_BLK          16
#define NBLK           (T_LEN / L_BLK)          // 256
#define NCHAN          32
#define NBATCH         64
#define SEQ_PER_BAND   (NBATCH * NCHAN)         // 2048
#define WAVES_PER_WG   8
#define WG_PER_BAND    (SEQ_PER_BAND / 16 / WAVES_PER_WG)  // 16

__device__ __forceinline__ v8f wmma4(v2f a, v2f b, v8f c) {
    // D(16x16,f32) = A(16x4,f32) * B(4x16,f32) + C
    return __builtin_amdgcn_wmma_f32_16x16x4_f32(
        /*neg_a=*/false, a, /*neg_b=*/false, b,
        /*c_mod=*/(short)0, c, /*reuse_a=*/false, /*reuse_b=*/false);
}

// FWD=true : src = x (layout seq*4096), dst = out (final layout), forward time.
// FWD=false: src = dst = out, reversed time, in-place (wave owns its columns).
template <bool FWD>
__global__ __launch_bounds__(256) void filtfilt_pass(
    const float* __restrict__ src_base,
    const float* __restrict__ Bco,     // (9,9)
    const float* __restrict__ Aco,     // (9,9)
    float* __restrict__ dst_base)
{
    const int band = blockIdx.x / WG_PER_BAND;
    const int tid  = threadIdx.x;

    // ------------------------------------------------------------------
    // Build M(24x24) for this band in LDS: column j = 16-step response to
    // basis input (j<8: z0=e_j; j>=8: unit impulse at t=j-8).
    // Rows 0..15 = y[t], rows 16..23 = z_16[i].
    // ------------------------------------------------------------------
    __shared__ float M[24][25];   // +1 pad vs bank conflicts
    if (tid < 24) {
        float b[NCOEF], a[NCOEF];
        const float a0 = Aco[band * NCOEF];
        #pragma unroll
        for (int k = 0; k < NCOEF; ++k) {
            b[k] = Bco[band * NCOEF + k] / a0;
            a[k] = Aco[band * NCOEF + k] / a0;
        }
        float z[ORDER + 1];
        #pragma unroll
        for (int i = 0; i <= ORDER; ++i) z[i] = 0.0f;
        if (tid < ORDER) z[tid] = 1.0f;
        #pragma unroll
        for (int t = 0; t < L_BLK; ++t) {
            const float xt = (tid >= 8 && (tid - 8) == t) ? 1.0f : 0.0f;
            const float y  = fmaf(b[0], xt, z[0]);
            #pragma unroll
            for (int i = 0; i < ORDER; ++i)          // z[8] stays 0
                z[i] = z[i + 1] + b[i + 1] * xt - a[i + 1] * y;
            M[t][tid] = y;
        }
        #pragma unroll
        for (int i = 0; i < ORDER; ++i) M[16 + i][tid] = z[i];
    }
    __syncthreads();

    const int lane = tid & 31;
    const int col  = lane & 15;     // output column N / matrix row m
    const int hi   = lane >> 4;     // lane group

    // ------------------------------------------------------------------
    // A operands (constant per band). ISA A 16x4 f32 layout:
    //   element (m = lane%16, kk = (lane/16)*2 + v) in VGPR v.
    // Slice s covers K-columns 4s..4s+3 of M.
    //   Ay: rows 0..15 of M (the Y block);  Az: rows 16..23 (Z'), rows 8..15 = 0.
    // ------------------------------------------------------------------
    v2f Ay[6], Az[6];
    #pragma unroll
    for (int s = 0; s < 6; ++s) {
        const int c0 = 4 * s + hi * 2;
        Ay[s].x = M[col][c0];
        Ay[s].y = M[col][c0 + 1];
        Az[s].x = (col < 8) ? M[16 + col][c0]     : 0.0f;
        Az[s].y = (col < 8) ? M[16 + col][c0 + 1] : 0.0f;
    }

    // This wave's 16 sequences.
    const int wave = tid >> 5;
    const int g    = (blockIdx.x % WG_PER_BAND) * WAVES_PER_WG + wave; // [0,128)
    const int seq  = g * 16 + col;                                     // [0,2048)
    const int batch = seq >> 5, ch = seq & 31;
    const size_t outoff = ((size_t)(batch * BANDS + band) * NCHAN + ch) * T_LEN;

    const float* __restrict__ src = FWD ? (src_base + (size_t)seq * T_LEN)
                                        : (src_base + outoff);
    float* __restrict__ dst = dst_base + outoff;

    v8f zacc = {0.f, 0.f, 0.f, 0.f, 0.f, 0.f, 0.f, 0.f};  // z0 = 0 (both passes)

    for (int blk = 0; blk < NBLK; ++blk) {
        const int t0 = blk * L_BLK;

        // ---- X in B-operand layout (4x16 f32 slices):
        //   slice s, VGPR v: lanes0-15 -> K=4s+v? No: lanes0-15 K=4s+{0,1} across
        //   V0,V1; lanes16-31 K=4s+{2,3}.  So this lane needs rows r, r+1 with
        //   r = 4s + 2*hi  -> one b64 load (consecutive t).
        v2f Bx[4];
        #pragma unroll
        for (int s = 0; s < 4; ++s) {
            const int r = 4 * s + 2 * hi;
            if (FWD) {
                Bx[s].x = src[t0 + r];
                Bx[s].y = src[t0 + r + 1];
            } else {
                Bx[s].x = src[T_LEN - 1 - (t0 + r)];
                Bx[s].y = src[T_LEN - 2 - (t0 + r)];
            }
        }

        // ---- Z (D layout: VGPR i = rows i / i+8) -> B layout, via shuffles.
        const float zr2 = __shfl(zacc[2], col, 32);
        const float zr3 = __shfl(zacc[3], col, 32);
        const float zr6 = __shfl(zacc[6], col, 32);
        const float zr7 = __shfl(zacc[7], col, 32);
        v2f Bz0, Bz1;
        Bz0.x = hi ? zr2 : zacc[0];
        Bz0.y = hi ? zr3 : zacc[1];
        Bz1.x = hi ? zr6 : zacc[4];
        Bz1.y = hi ? zr7 : zacc[5];

        // ---- Contract. X slices first (no cross-block dependence), state last.
        v8f yacc = {0.f, 0.f, 0.f, 0.f, 0.f, 0.f, 0.f, 0.f};
        v8f znew = {0.f, 0.f, 0.f, 0.f, 0.f, 0.f, 0.f, 0.f};
        #pragma unroll
        for (int s = 0; s < 4; ++s) {
            yacc = wmma4(Ay[2 + s], Bx[s], yacc);
            znew = wmma4(Az[2 + s], Bx[s], znew);
        }
        yacc = wmma4(Ay[0], Bz0, yacc);
        yacc = wmma4(Ay[1], Bz1, yacc);
        znew = wmma4(Az[0], Bz0, znew);   // serial cross-block chain: these two
        znew = wmma4(Az[1], Bz1, znew);

        // ---- Store Y (D layout: VGPR i = time t0 + i + 8*hi).
        if (FWD) {
            const int t = t0 + 8 * hi;
            float4 lo4 = make_float4(yacc[0], yacc[1], yacc[2], yacc[3]);
            float4 hi4 = make_float4(yacc[4], yacc[5], yacc[6], yacc[7]);
            *reinterpret_cast<float4*>(dst + t)     = lo4;
            *reinterpret_cast<float4*>(dst + t + 4) = hi4;
        } else {
            // reversed-time row m lands at final index T-1-(t0+m)
            const int base = T_LEN - 1 - t0 - 8 * hi;
            float4 lo4 = make_float4(yacc[3], yacc[2], yacc[1], yacc[0]);
            float4 hi4 = make_float4(yacc[7], yacc[6], yacc[5], yacc[4]);
            *reinterpret_cast<float4*>(dst + base - 3) = lo4;   // indices base-3..base
            *reinterpret_cast<float4*>(dst + base - 7) = hi4;   // indices base-7..base-4
        }

        zacc = znew;
    }
}

extern "C" void kernel_launch(void* const* d_in, const int* in_sizes, int n_in,
                              void* d_out, int out_size, void* d_ws, size_t ws_size,
                              hipStream_t stream) {
    (void)in_sizes; (void)n_in; (void)out_size; (void)d_ws; (void)ws_size;
    const float* x  = (const float*)d_in[0];   // (64,32,4096)
    const float* Bc = (const float*)d_in[1];   // (9,9)
    const float* Ac = (const float*)d_in[2];   // (9,9)
    float* out = (float*)d_out;                // (64,9,32,4096)

    dim3 grid(BANDS * WG_PER_BAND);            // 144 workgroups
    dim3 block(256);                           // 8 wave32 groups per WG
    // Pass 1: forward filter x -> out (already in final layout).
    filtfilt_pass<true><<<grid, block, 0, stream>>>(x, Bc, Ac, out);
    // Pass 2: backward filter in-place on out (each wave owns its columns).
    filtfilt_pass<false><<<grid, block, 0, stream>>>(out, Bc, Ac, out);
}